// SolarGCN_48653389529561
// MI455X (gfx1250) — compile-verified
//
#include <hip/hip_runtime.h>

#define NODES 10000
#define EDGES 320000
#define BN_EPS 1e-5f
#define GEMM_WAVES 5  // 625 row tiles = 5 waves/block * 125 blocks

typedef __attribute__((ext_vector_type(2))) float v2f;
typedef __attribute__((ext_vector_type(8))) float v8f;

// ---------------------------------------------------------------------------
// Degree / normalization kernels
// ---------------------------------------------------------------------------
__global__ void k_init_deg(float* __restrict__ deg, int n) {
  int i = blockIdx.x * blockDim.x + threadIdx.x;
  if (i < n) deg[i] = 1.0f;  // self-loop fill value
}

__global__ void k_deg_accum(const int* __restrict__ dst, const float* __restrict__ ew,
                            float* __restrict__ deg, int e) {
  int i = blockIdx.x * blockDim.x + threadIdx.x;
  if (i < e) atomicAdd(&deg[dst[i]], ew[i]);
}

__global__ void k_deg_finish(const float* __restrict__ deg, float* __restrict__ dis,
                             float* __restrict__ inv_deg, int n) {
  int i = blockIdx.x * blockDim.x + threadIdx.x;
  if (i < n) {
    float d = deg[i];
    dis[i] = rsqrtf(d);
    inv_deg[i] = 1.0f / d;
  }
}

__global__ void k_norm_e(const int* __restrict__ src, const int* __restrict__ dst,
                         const float* __restrict__ ew, const float* __restrict__ dis,
                         float* __restrict__ norm_e, int e) {
  int i = blockIdx.x * blockDim.x + threadIdx.x;
  if (i < e) norm_e[i] = dis[src[i]] * ew[i] * dis[dst[i]];
}

// ---------------------------------------------------------------------------
// WMMA fp32 GEMM: C[NxM] = A[NxK] @ B[KxM], optional fused agg epilogue:
//   agg = C * inv_deg[row] + bias[col]
//
// Block = 5 waves (160 threads). grid = (N/80, M/64).
// The K x 64 weight tile is staged once per block into LDS, swizzled so that
// a lane's two K-values (k, k+1) sit adjacent -> single ds_load_b64 per
// B fragment, conflict-free.
// Each wave computes a 16x64 strip: 4 v8f accumulators, one A fragment feeds
// 4 independent v_wmma_f32_16x16x4_f32 per k-step.
// N multiple of 16, K multiple of 4, M multiple of 64 -> no bounds checks,
// EXEC all-1s everywhere (WMMA requirement).
// ---------------------------------------------------------------------------
__global__ void __launch_bounds__(GEMM_WAVES * 32)
k_gemm_wmma_f32(const float* __restrict__ A, const float* __restrict__ B,
                float* __restrict__ C, float* __restrict__ agg,
                const float* __restrict__ inv_deg, const float* __restrict__ bias,
                int K, int M) {
  __shared__ float sB[256 * 64];  // up to K=256, 64 cols (64 KB)

  const int col_base = blockIdx.y * 64;
  const int tid = threadIdx.x;

  // Stage weight tile into LDS, swizzled: sB[(k>>1)*128 + c*2 + (k&1)]
  for (int i = tid; i < K * 64; i += GEMM_WAVES * 32) {
    int k = i >> 6, c = i & 63;
    sB[(k >> 1) * 128 + c * 2 + (k & 1)] = B[(size_t)k * M + col_base + c];
  }
  __syncthreads();

  const int wave = tid >> 5;
  const int row0 = (blockIdx.x * GEMM_WAVES + wave) << 4;
  const int lane = tid & 31;
  const int half = lane >> 4;  // 0: K pair {0,1}, 1: K pair {2,3}
  const int l16  = lane & 15;
  const int kb   = half << 1;

  v8f acc[4] = {v8f{}, v8f{}, v8f{}, v8f{}};
  const float* arow = A + (size_t)(row0 + l16) * K;

  for (int k0 = 0; k0 < K; k0 += 4) {
    const float* ap = arow + k0 + kb;
    v2f a; a.x = ap[0]; a.y = ap[1];
    const float* bl = &sB[((k0 + kb) >> 1) * 128];  // pair row in LDS
    v2f b0 = *(const v2f*)(bl + (l16 + 0)  * 2);
    v2f b1 = *(const v2f*)(bl + (l16 + 16) * 2);
    v2f b2 = *(const v2f*)(bl + (l16 + 32) * 2);
    v2f b3 = *(const v2f*)(bl + (l16 + 48) * 2);
    acc[0] = __builtin_amdgcn_wmma_f32_16x16x4_f32(false, a, false, b0, (short)0, acc[0], false, false);
    acc[1] = __builtin_amdgcn_wmma_f32_16x16x4_f32(false, a, false, b1, (short)0, acc[1], false, false);
    acc[2] = __builtin_amdgcn_wmma_f32_16x16x4_f32(false, a, false, b2, (short)0, acc[2], false, false);
    acc[3] = __builtin_amdgcn_wmma_f32_16x16x4_f32(false, a, false, b3, (short)0, acc[3], false, false);
  }

  // Epilogue: C VGPR i holds row (row0 + half*8 + i), col = tile_col + l16.
  const int mrow = half << 3;
  float idg[8];
  if (agg) {
#pragma unroll
    for (int i = 0; i < 8; ++i) idg[i] = inv_deg[row0 + mrow + i];
  }
#pragma unroll
  for (int j = 0; j < 4; ++j) {
    const int col = col_base + j * 16 + l16;
    float* cp = C + (size_t)(row0 + mrow) * M + col;
    float bj = agg ? bias[col] : 0.0f;
#pragma unroll
    for (int i = 0; i < 8; ++i) {
      float v = acc[j][i];
      cp[(size_t)i * M] = v;
      if (agg) agg[(size_t)(row0 + mrow + i) * M + col] = v * idg[i] + bj;
    }
  }
}

// ---------------------------------------------------------------------------
// Edge aggregation: agg[dst] += norm_e * h[src]  (float4 gather, 4 f32 atomics)
// Consecutive lanes cover consecutive feature chunks of one edge -> coalesced
// gather; scatter-adds resolve in L2 (whole h fits in the 192MB L2).
// ---------------------------------------------------------------------------
__global__ void k_edge_agg(const int* __restrict__ src, const int* __restrict__ dst,
                           const float* __restrict__ norm_e, const float* __restrict__ h,
                           float* __restrict__ agg, int M) {
  int per_edge = M >> 2;  // float4 chunks per edge
  int idx = blockIdx.x * blockDim.x + threadIdx.x;
  int total = EDGES * per_edge;
  if (idx >= total) return;
  int e = idx / per_edge;
  int f = (idx - e * per_edge) << 2;
  float w = norm_e[e];
  const float4 v = *(const float4*)(h + (size_t)src[e] * M + f);
  float* ap = agg + (size_t)dst[e] * M + f;
  atomicAdd(ap + 0, w * v.x);
  atomicAdd(ap + 1, w * v.y);
  atomicAdd(ap + 2, w * v.z);
  atomicAdd(ap + 3, w * v.w);
}

// ---------------------------------------------------------------------------
// Fused BN (+ optional skip [+ skip bias]) + ReLU
//   mode 0: relu(bn(agg) + skip + skb[col])   (projected skip, bias folded)
//   mode 1: relu(bn(agg) + skip)              (identity skip)
//   mode 2: relu(bn(agg))
// ---------------------------------------------------------------------------
__global__ void k_post(const float* __restrict__ agg, const float* __restrict__ skip,
                       const float* __restrict__ skb,
                       const float* __restrict__ g, const float* __restrict__ be,
                       const float* __restrict__ rm, const float* __restrict__ rv,
                       float* __restrict__ out, int M, int mode) {
  int idx = blockIdx.x * blockDim.x + threadIdx.x;
  int total = NODES * M;
  if (idx >= total) return;
  int col = idx % M;
  float scale = g[col] * rsqrtf(rv[col] + BN_EPS);
  float v = (agg[idx] - rm[col]) * scale + be[col];
  if (mode == 0) v += skip[idx] + skb[col];
  else if (mode == 1) v += skip[idx];
  out[idx] = v > 0.0f ? v : 0.0f;
}

// ---------------------------------------------------------------------------
// Final fc: out[i, 0:2] = h[i, 0:64] @ fcW[64x2] + fcb
// ---------------------------------------------------------------------------
__global__ void k_fc(const float* __restrict__ h, const float* __restrict__ fcW,
                     const float* __restrict__ fcb, float* __restrict__ out) {
  int i = blockIdx.x * blockDim.x + threadIdx.x;
  if (i >= NODES) return;
  const float* hp = h + (size_t)i * 64;
  float a0 = fcb[0], a1 = fcb[1];
#pragma unroll 8
  for (int f = 0; f < 64; ++f) {
    float x = hp[f];
    a0 += x * fcW[f * 2 + 0];
    a1 += x * fcW[f * 2 + 1];
  }
  out[i * 2 + 0] = a0;
  out[i * 2 + 1] = a1;
}

// ---------------------------------------------------------------------------
// Host orchestration
// ---------------------------------------------------------------------------
static inline int cdiv(int a, int b) { return (a + b - 1) / b; }

static void launch_gemm(const float* A, const float* B, float* C, float* agg,
                        const float* inv_deg, const float* bias,
                        int K, int M, hipStream_t stream) {
  dim3 grid(625 / GEMM_WAVES, M / 64);  // 125 row blocks x column groups
  k_gemm_wmma_f32<<<grid, GEMM_WAVES * 32, 0, stream>>>(A, B, C, agg, inv_deg, bias, K, M);
}

extern "C" void kernel_launch(void* const* d_in, const int* in_sizes, int n_in,
                              void* d_out, int out_size, void* d_ws, size_t ws_size,
                              hipStream_t stream) {
  const float* x    = (const float*)d_in[0];
  const int*   ei   = (const int*)d_in[1];  // [2, E]
  const float* ew   = (const float*)d_in[2];
  const float* W1   = (const float*)d_in[3];
  const float* b1   = (const float*)d_in[4];
  const float* g1   = (const float*)d_in[5];
  const float* be1  = (const float*)d_in[6];
  const float* rm1  = (const float*)d_in[7];
  const float* rv1  = (const float*)d_in[8];
  const float* sk1W = (const float*)d_in[9];
  const float* sk1b = (const float*)d_in[10];
  const float* W2   = (const float*)d_in[11];
  const float* b2   = (const float*)d_in[12];
  const float* g2   = (const float*)d_in[13];
  const float* be2  = (const float*)d_in[14];
  const float* rm2  = (const float*)d_in[15];
  const float* rv2  = (const float*)d_in[16];
  const float* W3   = (const float*)d_in[17];
  const float* b3   = (const float*)d_in[18];
  const float* g3   = (const float*)d_in[19];
  const float* be3  = (const float*)d_in[20];
  const float* rm3  = (const float*)d_in[21];
  const float* rv3  = (const float*)d_in[22];
  const float* sk2W = (const float*)d_in[23];
  const float* sk2b = (const float*)d_in[24];
  const float* W4   = (const float*)d_in[25];
  const float* b4   = (const float*)d_in[26];
  const float* g4   = (const float*)d_in[27];
  const float* be4  = (const float*)d_in[28];
  const float* rm4  = (const float*)d_in[29];
  const float* rv4  = (const float*)d_in[30];
  const float* fcW  = (const float*)d_in[31];
  const float* fcb  = (const float*)d_in[32];

  const int* src = ei;          // edge_index[0]
  const int* dst = ei + EDGES;  // edge_index[1]

  // Workspace layout (floats)
  float* ws = (float*)d_ws;
  float* deg     = ws;                   // [N]
  float* dis     = deg + NODES;          // [N]
  float* inv_deg = dis + NODES;          // [N]
  float* norm_e  = inv_deg + NODES;      // [E]
  float* B0 = norm_e + EDGES;            // [N*256] GEMM output h
  float* B1 = B0 + (size_t)NODES * 256;  // [N*256] agg
  float* B2 = B1 + (size_t)NODES * 256;  // [N*256] skip proj / final h4
  float* B3 = B2 + (size_t)NODES * 256;  // [N*256] id2 (layer1 out)
  float* B4 = B3 + (size_t)NODES * 256;  // [N*256] layer2 / layer3 out

  const int T = 256;

  // --- normalization terms ---
  k_init_deg<<<cdiv(NODES, T), T, 0, stream>>>(deg, NODES);
  k_deg_accum<<<cdiv(EDGES, T), T, 0, stream>>>(dst, ew, deg, EDGES);
  k_deg_finish<<<cdiv(NODES, T), T, 0, stream>>>(deg, dis, inv_deg, NODES);
  k_norm_e<<<cdiv(EDGES, T), T, 0, stream>>>(src, dst, ew, dis, norm_e, EDGES);

  // --- layer 1: 128 -> 256, projected skip from x ---
  launch_gemm(x, W1, B0, B1, inv_deg, b1, 128, 256, stream);  // h + fused agg init
  k_edge_agg<<<cdiv(EDGES * 64, T), T, 0, stream>>>(src, dst, norm_e, B0, B1, 256);
  launch_gemm(x, sk1W, B2, nullptr, nullptr, nullptr, 128, 256, stream);
  k_post<<<cdiv(NODES * 256, T), T, 0, stream>>>(B1, B2, sk1b, g1, be1, rm1, rv1, B3, 256, 0);

  // --- layer 2: 256 -> 256, identity skip (id2 = B3) ---
  launch_gemm(B3, W2, B0, B1, inv_deg, b2, 256, 256, stream);
  k_edge_agg<<<cdiv(EDGES * 64, T), T, 0, stream>>>(src, dst, norm_e, B0, B1, 256);
  k_post<<<cdiv(NODES * 256, T), T, 0, stream>>>(B1, B3, nullptr, g2, be2, rm2, rv2, B4, 256, 1);

  // --- layer 3: 256 -> 128, projected skip from id2 ---
  launch_gemm(B4, W3, B0, B1, inv_deg, b3, 256, 128, stream);
  k_edge_agg<<<cdiv(EDGES * 32, T), T, 0, stream>>>(src, dst, norm_e, B0, B1, 128);
  launch_gemm(B3, sk2W, B2, nullptr, nullptr, nullptr, 256, 128, stream);
  k_post<<<cdiv(NODES * 128, T), T, 0, stream>>>(B1, B2, sk2b, g3, be3, rm3, rv3, B4, 128, 0);

  // --- layer 4: 128 -> 64, no skip ---
  launch_gemm(B4, W4, B0, B1, inv_deg, b4, 128, 64, stream);
  k_edge_agg<<<cdiv(EDGES * 16, T), T, 0, stream>>>(src, dst, norm_e, B0, B1, 64);
  k_post<<<cdiv(NODES * 64, T), T, 0, stream>>>(B1, nullptr, nullptr, g4, be4, rm4, rv4, B2, 64, 2);

  // --- fc: 64 -> 2 ---
  k_fc<<<cdiv(NODES, T), T, 0, stream>>>(B2, fcW, fcb, (float*)d_out);
}